// Aspect_Category_Prediction_v3_55654186222168
// MI455X (gfx1250) — compile-verified
//
#include <hip/hip_runtime.h>

#define INPUT_DIM  768
#define HIDDEN_DIM 256
#define ASPECT_NUM 20
#define BATCH      1024

typedef __attribute__((ext_vector_type(2))) float v2f;
typedef __attribute__((ext_vector_type(8))) float v8f;

#define XS_STRIDE 772   // 768 + 4 pad: row stride % 64 banks = 4 -> 16 rows hit 16 distinct banks
#define O1_STRIDE 260   // 256 + 4 pad

// ---------------- Kernel 1: bucket samples by aspect id ----------------
__global__ __launch_bounds__(1024) void group_by_aspect(const int* __restrict__ ids,
                                                        int* __restrict__ counts,
                                                        int* __restrict__ lists) {
    __shared__ int cnt[ASPECT_NUM];
    const int t = threadIdx.x;
    if (t < ASPECT_NUM) cnt[t] = 0;
    __syncthreads();
    if (t < BATCH) {
        const int a = ids[t];
        const int slot = atomicAdd(&cnt[a], 1);
        lists[a * BATCH + slot] = t;
    }
    __syncthreads();
    if (t < ASPECT_NUM) counts[t] = cnt[t];
}

// ---------------- Kernel 2: grouped fused 2-layer MLP via f32 WMMA ----------------
// grid = (BATCH/16, ASPECT_NUM); block = 512 threads (16 wave32s).
// Wave w computes out1[M=16 samples][N = w*16 .. w*16+15] over K=768 with
// V_WMMA_F32_16X16X4_F32, then layer 2 (256 -> 2) is a tiny LDS reduction.
__global__ __launch_bounds__(512) void aspect_mlp(const float* __restrict__ X,
                                                  const float* __restrict__ W1,
                                                  const float* __restrict__ b1,
                                                  const float* __restrict__ W2,
                                                  const float* __restrict__ b2,
                                                  const int* __restrict__ counts,
                                                  const int* __restrict__ lists,
                                                  float* __restrict__ out) {
    __shared__ float smem[16 * XS_STRIDE];   // X tile, later reused for out1 tile
    __shared__ int   rowidx[16];

    const int a    = blockIdx.y;
    const int tile = blockIdx.x;
    const int cnt  = counts[a];
    if (tile * 16 >= cnt) return;            // uniform per block: safe early exit

    const int tid  = threadIdx.x;
    const int wave = tid >> 5;
    const int lane = tid & 31;

    if (tid < 16) {
        const int p = tile * 16 + tid;
        rowidx[tid] = (p < cnt) ? lists[a * BATCH + p] : -1;
    }
    __syncthreads();

    // Stage 16 gathered X rows into LDS (float4), zeros for tail-masked rows.
    {
        const int nvec = 16 * (INPUT_DIM / 4);          // 3072 float4 chunks
        for (int i = tid; i < nvec; i += 512) {
            const int row = i / (INPUT_DIM / 4);
            const int c4  = i % (INPUT_DIM / 4);
            const int r   = rowidx[row];
            float4 v = make_float4(0.f, 0.f, 0.f, 0.f);
            if (r >= 0) v = *(const float4*)(X + (size_t)r * INPUT_DIM + c4 * 4);
            *(float4*)(&smem[row * XS_STRIDE + c4 * 4]) = v;
        }
    }
    __syncthreads();

    // ---- Layer 1: D = A(16xK) * B(Kx16) + b1, K stepped by 4 ----
    const int n0   = wave * 16;
    const int half = lane >> 4;       // 0: K+{0,1}, 1: K+{2,3}
    const int l16  = lane & 15;       // A: row M ; B: col N

    const float* xrow = &smem[l16 * XS_STRIDE + half * 2];
    const float* w1p  = W1 + (size_t)a * (INPUT_DIM * HIDDEN_DIM)
                           + (size_t)(half * 2) * HIDDEN_DIM + n0 + l16;

    const float bval = b1[a * HIDDEN_DIM + n0 + l16];   // depends on N only
    v8f acc = {bval, bval, bval, bval, bval, bval, bval, bval};

    #pragma unroll 4
    for (int k0 = 0; k0 < INPUT_DIM; k0 += 4) {
        v2f af = *(const v2f*)(xrow + k0);              // A: X[m][k0+2h .. +1] (b64 LDS load)
        v2f bf;
        bf.x = w1p[(size_t)k0 * HIDDEN_DIM];            // B: W1[k0+2h  ][n] (L2-resident)
        bf.y = w1p[(size_t)k0 * HIDDEN_DIM + HIDDEN_DIM]; //  W1[k0+2h+1][n]
        acc = __builtin_amdgcn_wmma_f32_16x16x4_f32(
            /*neg_a=*/false, af, /*neg_b=*/false, bf,
            /*c_mod=*/(short)0, acc, /*reuse_a=*/false, /*reuse_b=*/false);
    }

    __syncthreads();   // everyone done reading the X tile; reuse LDS for out1

    float* out1 = smem;
    #pragma unroll
    for (int v = 0; v < 8; ++v) {
        const int m = v + 8 * half;                     // C/D layout: vgpr v, lane half
        const float val = acc[v];
        out1[m * O1_STRIDE + n0 + l16] = val > 0.f ? val : 0.f;   // ReLU
    }
    __syncthreads();

    // ---- Layer 2: logits = out1(16x256) * W2(256x2) + b2 ----
    if (tid < 32) {
        const int m   = tid & 15;
        const int col = tid >> 4;
        if (tile * 16 + m < cnt) {
            const int r = rowidx[m];
            const float* w2p = W2 + (size_t)a * (HIDDEN_DIM * 2) + col;
            const float* o1  = &out1[m * O1_STRIDE];
            float s = b2[a * 2 + col];
            #pragma unroll 8
            for (int h = 0; h < HIDDEN_DIM; ++h) s += o1[h] * w2p[2 * h];
            out[(size_t)r * 2 + col] = s;
        }
    }
}

extern "C" void kernel_launch(void* const* d_in, const int* in_sizes, int n_in,
                              void* d_out, int out_size, void* d_ws, size_t ws_size,
                              hipStream_t stream) {
    (void)in_sizes; (void)n_in; (void)out_size; (void)ws_size;
    const float* X   = (const float*)d_in[0];
    const int*   ids = (const int*)  d_in[1];
    const float* W1  = (const float*)d_in[2];
    const float* b1  = (const float*)d_in[3];
    const float* W2  = (const float*)d_in[4];
    const float* b2  = (const float*)d_in[5];
    float* out = (float*)d_out;

    int* counts = (int*)d_ws;                 // ASPECT_NUM ints
    int* lists  = counts + ASPECT_NUM;        // ASPECT_NUM * BATCH ints (~82 KB total)

    group_by_aspect<<<1, 1024, 0, stream>>>(ids, counts, lists);

    dim3 grid(BATCH / 16, ASPECT_NUM);        // 64 tiles worst-case per aspect; empty tiles exit
    aspect_mlp<<<grid, 512, 0, stream>>>(X, W1, b1, W2, b2, counts, lists, out);
}